// BodhiTTS_23587960390224
// MI455X (gfx1250) — compile-verified
//
#include <hip/hip_runtime.h>

#define B_  32
#define TT  512
#define TM  2048
#define DD  512
#define MM  80
#define NEGV (-1e9f)

typedef __attribute__((ext_vector_type(2))) float v2f;
typedef __attribute__((ext_vector_type(8))) float v8f;

__device__ __forceinline__ unsigned ballot32(bool p) {
#if __has_builtin(__builtin_amdgcn_ballot_w32)
  return __builtin_amdgcn_ballot_w32(p);
#else
  return (unsigned)__ballot(p);
#endif
}

__device__ __forceinline__ v8f v8zero() {
  v8f z;
#pragma unroll
  for (int q = 0; q < 8; ++q) z[q] = 0.0f;
  return z;
}

// ---------------------------------------------------------------------------
// Kernel 1: Hw[b,i,m] = sum_d h_text[b,i,d] * w_proj[d,m]   (WMMA f32 16x16x4)
//           hb[b,i]   = sum_d h_text[b,i,d] * b_proj[d]     (VALU side-band)
// Wave tile: 32(i) x 80(m) = 2x5 WMMA tiles, K=512.
// ---------------------------------------------------------------------------
__global__ void k_hw_wmma(const float* __restrict__ h_text,
                          const float* __restrict__ w_proj,
                          const float* __restrict__ b_proj,
                          float* __restrict__ Hw,
                          float* __restrict__ hb) {
  const int b    = blockIdx.y;
  const int wave = threadIdx.x >> 5;
  const int lane = threadIdx.x & 31;
  const int r    = lane & 15;
  const int h    = lane >> 4;
  const int i0   = (blockIdx.x * 4 + wave) * 32;

  const float* Abase = h_text + ((size_t)b * TT + i0) * DD;

  v8f acc[10];
#pragma unroll
  for (int q = 0; q < 10; ++q) acc[q] = v8zero();
  float pb0 = 0.0f, pb1 = 0.0f;

  for (int k = 0; k < DD; k += 4) {
    v2f a[2];
#pragma unroll
    for (int mt = 0; mt < 2; ++mt)
      a[mt] = *(const v2f*)(Abase + (size_t)(mt * 16 + r) * DD + k + 2 * h);

    v2f bp = *(const v2f*)(b_proj + k + 2 * h);
    pb0 = fmaf(a[0][0], bp[0], fmaf(a[0][1], bp[1], pb0));
    pb1 = fmaf(a[1][0], bp[0], fmaf(a[1][1], bp[1], pb1));

    v2f bf[5];
#pragma unroll
    for (int nt = 0; nt < 5; ++nt) {
      const float* wp = w_proj + (size_t)(k + 2 * h) * MM + nt * 16 + r;
      bf[nt][0] = wp[0];
      bf[nt][1] = wp[MM];
    }
#pragma unroll
    for (int mt = 0; mt < 2; ++mt)
#pragma unroll
      for (int nt = 0; nt < 5; ++nt)
        acc[mt * 5 + nt] = __builtin_amdgcn_wmma_f32_16x16x4_f32(
            false, a[mt], false, bf[nt], (short)0, acc[mt * 5 + nt], false, false);
  }

  // hb: combine the two K-halves held by lane and lane^16
  float t0 = pb0 + __shfl_xor(pb0, 16);
  float t1 = pb1 + __shfl_xor(pb1, 16);
  if (h == 0) {
    hb[(size_t)b * TT + i0 + r]      = t0;
    hb[(size_t)b * TT + i0 + 16 + r] = t1;
  }

#pragma unroll
  for (int mt = 0; mt < 2; ++mt)
#pragma unroll
    for (int nt = 0; nt < 5; ++nt)
#pragma unroll
      for (int v = 0; v < 8; ++v) {
        int ii = i0 + mt * 16 + v + 8 * h;
        Hw[((size_t)b * TT + ii) * MM + nt * 16 + r] = acc[mt * 5 + nt][v];
      }
}

// ---------------------------------------------------------------------------
// Kernel 2: attn[b,j,i] = sum_m Hw[b,i,m] * mel[b,m,j] + hb[b,i], masked NEG.
// Stored TRANSPOSED [B, Tm, Tt] so softmax/DP read contiguous rows.
// Wave tile: 64(j) x 32(i) = 4x2 WMMA tiles, K=80.
// ---------------------------------------------------------------------------
__global__ void k_attn_wmma(const float* __restrict__ Hw,
                            const float* __restrict__ mel,
                            const float* __restrict__ hb,
                            const int* __restrict__ text_len,
                            const int* __restrict__ mel_len,
                            float* __restrict__ attn) {
  const int b    = blockIdx.z;
  const int wave = threadIdx.x >> 5;
  const int lane = threadIdx.x & 31;
  const int r    = lane & 15;
  const int h    = lane >> 4;
  const int i0   = (blockIdx.x * 4 + wave) * 32;  // N dim (text)
  const int j0   = blockIdx.y * 64;               // M dim (mel frame)

  const float* melb = mel + (size_t)b * MM * TM;
  const float* Hwb  = Hw + (size_t)b * TT * MM;

  v8f acc[8];
#pragma unroll
  for (int q = 0; q < 8; ++q) acc[q] = v8zero();

  for (int k = 0; k < MM; k += 4) {
    v2f a[4];
#pragma unroll
    for (int mt = 0; mt < 4; ++mt) {
      const float* mp = melb + (size_t)(k + 2 * h) * TM + j0 + mt * 16 + r;
      a[mt][0] = mp[0];
      a[mt][1] = mp[TM];
    }
    v2f bf[2];
#pragma unroll
    for (int nt = 0; nt < 2; ++nt)
      bf[nt] = *(const v2f*)(Hwb + (size_t)(i0 + nt * 16 + r) * MM + k + 2 * h);
#pragma unroll
    for (int mt = 0; mt < 4; ++mt)
#pragma unroll
      for (int nt = 0; nt < 2; ++nt)
        acc[mt * 2 + nt] = __builtin_amdgcn_wmma_f32_16x16x4_f32(
            false, a[mt], false, bf[nt], (short)0, acc[mt * 2 + nt], false, false);
  }

  const int tl = text_len[b];
  const int ml = mel_len[b];
#pragma unroll
  for (int nt = 0; nt < 2; ++nt) {
    const int col = i0 + nt * 16 + r;
    const float bias = hb[(size_t)b * TT + col];
    const bool colok = col < tl;
#pragma unroll
    for (int mt = 0; mt < 4; ++mt)
#pragma unroll
      for (int v = 0; v < 8; ++v) {
        int row = j0 + mt * 16 + v + 8 * h;
        float val = acc[mt * 2 + nt][v] + bias;
        attn[((size_t)b * TM + row) * TT + col] = (colok && row < ml) ? val : NEGV;
      }
  }
}

// ---------------------------------------------------------------------------
// Kernel 3: in-place masked log_softmax over text axis (contiguous rows).
// ---------------------------------------------------------------------------
__global__ void k_logsoftmax(float* __restrict__ attn,
                             const int* __restrict__ text_len,
                             const int* __restrict__ mel_len) {
  const int b = blockIdx.y, j = blockIdx.x;
  if (j >= mel_len[b]) return;  // unused by DP/backtrack
  float* row = attn + ((size_t)b * TM + j) * TT;
  const int tid = threadIdx.x;
  __shared__ float red[256];

  float v0 = row[tid], v1 = row[tid + 256];
  float m = fmaxf(v0, v1);
  red[tid] = m; __syncthreads();
  for (int s = 128; s > 0; s >>= 1) {
    if (tid < s) red[tid] = fmaxf(red[tid], red[tid + s]);
    __syncthreads();
  }
  m = red[0]; __syncthreads();
  red[tid] = expf(v0 - m) + expf(v1 - m); __syncthreads();
  for (int s = 128; s > 0; s >>= 1) {
    if (tid < s) red[tid] += red[tid + s];
    __syncthreads();
  }
  float l = logf(red[0]);
  const int tl = text_len[b];
  row[tid]       = (tid < tl)       ? (v0 - m - l) : NEGV;
  row[tid + 256] = (tid + 256 < tl) ? (v1 - m - l) : NEGV;
}

// ---------------------------------------------------------------------------
// Kernel 4: MAS forward DP + LDS backtrack + durations + cumsum + regulate
// index + dur-loss partials. One 512-thread workgroup per batch.
// LDS: 128KB bit-packed choices + double-buffered Q line (1 barrier/frame).
// ---------------------------------------------------------------------------
__global__ void k_mas(const float* __restrict__ logp,
                      const int* __restrict__ text_len,
                      const int* __restrict__ mel_len,
                      const float* __restrict__ log_dur_pred,
                      float* __restrict__ dur_out,   // [B,TT] (float)
                      int* __restrict__ idx_ws,      // [B,TM]
                      float* __restrict__ partials)  // [2B]
{
  extern __shared__ unsigned char smemD[];
  unsigned* chbits = (unsigned*)smemD;            // TM*16 words (bit i per frame)
  float* qbuf  = (float*)(chbits + TM * 16);      // 2*TT
  int*   duri  = (int*)(qbuf + 2 * TT);           // TT
  int*   scanA = duri + TT;                       // TT
  int*   scanB = scanA + TT;                      // TT
  float* red   = (float*)(scanB + TT);            // TT

  const int b   = blockIdx.x;
  const int tid = threadIdx.x;
  const int tl  = text_len[b];
  const int ml  = mel_len[b];
  const float* base = logp + (size_t)b * TM * TT;

  if (tid < 16) chbits[tid] = 0u;
  duri[tid] = 0;

  // Q[i, j=0] = (i==0) ? logp[b,0,0] : NEG
  float qi = (tid == 0) ? base[0] : NEGV;

  for (int j = 1; j < ml; ++j) {
    float lp = base[(size_t)j * TT + tid];
    float* qb = qbuf + ((j & 1) * TT);
    qb[tid] = qi;
    __syncthreads();
    float qm1 = (tid == 0) ? NEGV : qb[tid - 1];
    bool take = qm1 > qi;                 // came from i-1
    qi = lp + fmaxf(qi, qm1);
    unsigned w = ballot32(take);
    if ((tid & 31) == 0) chbits[j * 16 + (tid >> 5)] = w;
  }
  __syncthreads();

  // backtrack entirely from LDS
  if (tid == 0) {
    int i = tl - 1;
    for (int j = ml - 1; j >= 0; --j) {
      duri[i] += 1;
      if (j > 0) {
        unsigned w = chbits[j * 16 + (i >> 5)];
        i -= (int)((w >> (i & 31)) & 1u);
        if (i < 0) i = 0;
      }
    }
  }
  __syncthreads();

  int dv = duri[tid];
  dur_out[(size_t)b * TT + tid] = (float)dv;

  // inclusive scan of durations (Hillis-Steele, double buffer)
  int* src = scanA; int* dst = scanB;
  src[tid] = dv; __syncthreads();
  for (int off = 1; off < TT; off <<= 1) {
    int x = src[tid];
    if (tid >= off) x += src[tid - off];
    dst[tid] = x;
    __syncthreads();
    int* tmp = src; src = dst; dst = tmp;
  }

  // length-regulate gather index: idx = #{i : cum[i] <= j}, clipped
  for (int jj = tid; jj < TM; jj += TT) {
    int lo = 0, hi = TT;
    while (lo < hi) {
      int mid = (lo + hi) >> 1;
      if (src[mid] <= jj) lo = mid + 1; else hi = mid;
    }
    idx_ws[(size_t)b * TM + jj] = (lo < TT - 1) ? lo : (TT - 1);
  }

  // masked dur-loss partials
  float dm = (tid < tl) ? 1.0f : 0.0f;
  float lg = logf(fmaxf((float)dv, 1.0f));
  float diff = (log_dur_pred[(size_t)b * TT + tid] - lg) * dm;
  float sq = diff * diff;
  red[tid] = sq; __syncthreads();
  for (int s = 256; s > 0; s >>= 1) {
    if (tid < s) red[tid] += red[tid + s];
    __syncthreads();
  }
  if (tid == 0) partials[2 * b] = red[0];
  __syncthreads();
  red[tid] = dm; __syncthreads();
  for (int s = 256; s > 0; s >>= 1) {
    if (tid < s) red[tid] += red[tid + s];
    __syncthreads();
  }
  if (tid == 0) partials[2 * b + 1] = red[0];
}

// ---------------------------------------------------------------------------
// Kernel 5: final loss reduction (deterministic, single thread over 32 pairs)
// ---------------------------------------------------------------------------
__global__ void k_loss(const float* __restrict__ partials, float* __restrict__ out) {
  if (threadIdx.x == 0) {
    float s = 0.0f, c = 0.0f;
    for (int b = 0; b < B_; ++b) { s += partials[2 * b]; c += partials[2 * b + 1]; }
    out[0] = s / c;
  }
}

// ---------------------------------------------------------------------------
// Kernel 6: length regulate — coalesced float4 row gather, masked zero
// ---------------------------------------------------------------------------
__global__ void k_regulate(const float* __restrict__ h_text,
                           const int* __restrict__ idx_ws,
                           const int* __restrict__ mel_len,
                           float* __restrict__ hexp) {
  const int b = blockIdx.y, j = blockIdx.x;
  const int tid = threadIdx.x;
  float4* dst = (float4*)(hexp + ((size_t)b * TM + j) * DD);
  if (j < mel_len[b]) {
    int idx = idx_ws[(size_t)b * TM + j];
    const float4* srcr = (const float4*)(h_text + ((size_t)b * TT + idx) * DD);
    dst[tid] = srcr[tid];
  } else {
    dst[tid] = make_float4(0.0f, 0.0f, 0.0f, 0.0f);
  }
}

// ---------------------------------------------------------------------------
extern "C" void kernel_launch(void* const* d_in, const int* in_sizes, int n_in,
                              void* d_out, int out_size, void* d_ws, size_t ws_size,
                              hipStream_t stream) {
  (void)in_sizes; (void)n_in; (void)out_size; (void)ws_size;

  const float* h_text   = (const float*)d_in[0];  // [B,TT,DD]
  const float* mel      = (const float*)d_in[1];  // [B,MM,TM]
  const int*   text_len = (const int*)d_in[2];    // [B]
  const int*   mel_len  = (const int*)d_in[3];    // [B]
  const float* w_proj   = (const float*)d_in[4];  // [DD,MM]
  const float* b_proj   = (const float*)d_in[5];  // [DD]
  const float* ldp      = (const float*)d_in[6];  // [B,TT]

  // workspace layout (~139.8 MB)
  float* attn     = (float*)d_ws;                          // B*TM*TT
  float* Hw       = attn + (size_t)B_ * TM * TT;           // B*TT*MM
  float* hb       = Hw + (size_t)B_ * TT * MM;             // B*TT
  int*   idx_ws   = (int*)(hb + (size_t)B_ * TT);          // B*TM
  float* partials = (float*)(idx_ws + (size_t)B_ * TM);    // 2*B

  // output layout: h_expanded | dur_loss | durations(float)
  float* hexp     = (float*)d_out;
  float* loss_out = hexp + (size_t)B_ * TM * DD;
  float* dur_out  = loss_out + 1;

  k_hw_wmma<<<dim3(TT / 128, B_), 128, 0, stream>>>(h_text, w_proj, b_proj, Hw, hb);
  k_attn_wmma<<<dim3(TT / 128, TM / 64, B_), 128, 0, stream>>>(Hw, mel, hb, text_len, mel_len, attn);
  k_logsoftmax<<<dim3(TM, B_), 256, 0, stream>>>(attn, text_len, mel_len);

  size_t smD = (size_t)(TM * 16 + 2 * TT + 4 * TT) * 4;  // 143,360 B < 320 KB
  k_mas<<<B_, TT, smD, stream>>>(attn, text_len, mel_len, ldp, dur_out, idx_ws, partials);
  k_loss<<<1, 32, 0, stream>>>(partials, loss_out);
  k_regulate<<<dim3(TM, B_), 128, 0, stream>>>(h_text, idx_ws, mel_len, hexp);
}